// YoloLoss_38070590112501
// MI455X (gfx1250) — compile-verified
//
#include <hip/hip_runtime.h>
#include <hip/hip_bf16.h>

// ---------------------------------------------------------------------------
// YOLO loss, MI455X (gfx1250, wave32).
// Bandwidth-bound (~183 MB traffic, ~8us at 23.3 TB/s). Strategy:
//  - stage per-cell data into LDS via gfx1250 async global->LDS b128 DMA
//    (cache->LDS direct, no VGPR round trip; ASYNCcnt + s_wait_asynccnt),
//    falling back to coalesced b128 loads where the builtin is absent
//  - one cell per thread out of LDS (conflict-free: stride 30/25 vs 32 lanes)
//  - deterministic per-block reduction (wave shfl butterfly + LDS) into d_ws
//  - final wave collapses lane partials with V_WMMA_F32_16X16X4_F32
//    (ones-matmul reduction, exact f32, deterministic).
// ---------------------------------------------------------------------------

typedef __attribute__((ext_vector_type(2))) float v2f;
typedef __attribute__((ext_vector_type(8))) float v8f;

#if __has_builtin(__builtin_amdgcn_global_load_async_to_lds_b128) && \
    __has_builtin(__builtin_amdgcn_s_wait_asynccnt)
#define USE_ASYNC_LDS 1
// Builtin signature (from toolchain diagnostic): 
//   void __builtin_amdgcn_global_load_async_to_lds_b128(
//        v4i AS(1)* src_global, v4i AS(3)* dst_lds, imm int offset, imm int cpol)
typedef __attribute__((__vector_size__(4 * sizeof(int)))) int v4i;
typedef __attribute__((address_space(1))) v4i* gv4i_p;   // pointer to global v4i
typedef __attribute__((address_space(3))) v4i* lv4i_p;   // pointer to LDS v4i
#else
#define USE_ASYNC_LDS 0
#endif

#define NCLS 20
#define CPB  256          // cells per block

__device__ __forceinline__ float4 to_xyxy(float x, float y, float w, float h,
                                          float fi, float fj) {
    // cx = CELL*(x+i), cy = CELL*(y+j), W = w*IMG, H = h*IMG  (CELL=64, IMG=448)
    float cx = 64.0f * (x + fi);
    float cy = 64.0f * (y + fj);
    float W  = w * 448.0f;
    float H  = h * 448.0f;
    float4 r;
    r.x = cx - 0.5f * W;
    r.y = cy - 0.5f * H;
    r.z = cx + 0.5f * W;
    r.w = cy + 0.5f * H;
    if (x == 0.0f && y == 0.0f && w == 0.0f && h == 0.0f) {
        r.x = 0.0f; r.y = 0.0f; r.z = 0.0f; r.w = 0.0f;
    }
    return r;
}

__device__ __forceinline__ float iou_of(const float4 gb, float area_g, const float4 pb) {
    float ltx = fmaxf(gb.x, pb.x);
    float lty = fmaxf(gb.y, pb.y);
    float rbx = fminf(gb.z, pb.z);
    float rby = fminf(gb.w, pb.w);
    float iw  = fmaxf(rbx - ltx, 0.0f);
    float ih  = fmaxf(rby - lty, 0.0f);
    float inter  = iw * ih;
    float area_p = (pb.z - pb.x) * (pb.w - pb.y);
    return inter / (area_g + area_p - inter);
}

__global__ __launch_bounds__(256) void yolo_cells(const float* __restrict__ pred,
                                                  const float* __restrict__ gt,
                                                  float* __restrict__ out,
                                                  float* __restrict__ blocksum,
                                                  int ncells)
{
    __shared__ float4 sp4[CPB * 30 / 4];   // 30720 B
    __shared__ float4 sg4[CPB * 25 / 4];   // 25600 B
    __shared__ float  wsum[8];             // one partial per wave32

    const int tid  = threadIdx.x;
    const int base = blockIdx.x * CPB;

    // Warm L2 with next block's chunk (global_prefetch_b8 path on gfx1250).
    if (blockIdx.x + 1 < gridDim.x) {
        __builtin_prefetch(pred + (size_t)(base + CPB) * 30 + (size_t)tid * 30, 0, 1);
        __builtin_prefetch(gt   + (size_t)(base + CPB) * 25 + (size_t)tid * 25, 0, 1);
    }

    // Stage chunk into LDS (16B-aligned: chunk strides 30720B / 25600B).
    const float4* p4 = (const float4*)(pred + (size_t)base * 30);
    const float4* g4 = (const float4*)(gt   + (size_t)base * 25);
#if USE_ASYNC_LDS
    for (int k = tid; k < CPB * 30 / 4; k += 256)
        __builtin_amdgcn_global_load_async_to_lds_b128(
            (gv4i_p)(v4i*)(p4 + k), (lv4i_p)(v4i*)(sp4 + k), 0, 0);
    for (int k = tid; k < CPB * 25 / 4; k += 256)
        __builtin_amdgcn_global_load_async_to_lds_b128(
            (gv4i_p)(v4i*)(g4 + k), (lv4i_p)(v4i*)(sg4 + k), 0, 0);
    __builtin_amdgcn_s_wait_asynccnt(0);
#else
    for (int k = tid; k < CPB * 30 / 4; k += 256) sp4[k] = p4[k];
    for (int k = tid; k < CPB * 25 / 4; k += 256) sg4[k] = g4[k];
#endif
    __syncthreads();

    const float* p = ((const float*)sp4) + tid * 30;
    const float* g = ((const float*)sg4) + tid * 25;
    const int cid  = base + tid;

    float cls = 0.0f;
    if (cid < ncells) {
        const float mask = (g[NCLS] == 1.0f) ? 1.0f : 0.0f;

        // class loss partial: mask * sum_c (p_c - g_c)^2
        #pragma unroll
        for (int c = 0; c < NCLS; ++c) {
            float d = p[c] - g[c];
            cls += d * d;
        }
        cls *= mask;

        // grid coordinates: cid = (b*7 + i)*7 + j ; cx uses i (axis 1), cy uses j (axis 2)
        const int j = cid % 7;
        const int i = (cid / 7) % 7;
        const float fi = (float)i;
        const float fj = (float)j;

        // gt box: [obj, x, y, w, h] at g[20..24]
        const float4 gb = to_xyxy(g[21], g[22], g[23], g[24], fi, fj);
        const float area_g = (gb.z - gb.x) * (gb.w - gb.y);

        // pred boxes: p[20..24] = [c,x,y,w,h] box0, p[25..29] box1
        const float4 pb0 = to_xyxy(p[21], p[22], p[23], p[24], fi, fj);
        const float4 pb1 = to_xyxy(p[26], p[27], p[28], p[29], fi, fj);
        const float iou0 = iou_of(gb, area_g, pb0);
        const float iou1 = iou_of(gb, area_g, pb1);

        // argmax (box0 wins ties, matching jnp.argmax first-max semantics)
        const int sel = (iou1 > iou0) ? 1 : 0;
        out[1 + (size_t)cid * 2 + 0] = (sel == 0) ? mask : 0.0f;
        out[1 + (size_t)cid * 2 + 1] = (sel == 1) ? mask : 0.0f;
    }

    // deterministic block reduction: wave32 butterfly, then 8 wave partials
    float v = cls;
    #pragma unroll
    for (int off = 16; off > 0; off >>= 1) v += __shfl_xor(v, off, 32);
    if ((tid & 31) == 0) wsum[tid >> 5] = v;
    __syncthreads();
    if (tid == 0) {
        float s = 0.0f;
        #pragma unroll
        for (int w = 0; w < 8; ++w) s += wsum[w];
        blocksum[blockIdx.x] = s;
    }
}

// One wave32: fold block partials (fixed order, b128 loads), then collapse
// the 32 lane partials with V_WMMA_F32_16X16X4_F32 against a ones B-matrix:
// D[i][j] = p_i + p_{i+16}. Exact f32, deterministic.
__global__ __launch_bounds__(32) void yolo_reduce(const float* __restrict__ blocksum,
                                                  int nblocks,
                                                  float* __restrict__ out,
                                                  float inv_batch)
{
    const int lane = threadIdx.x;
    float acc = 0.0f;
    const int n4 = nblocks >> 2;                 // full float4 groups
    const float4* bs4 = (const float4*)blocksum; // d_ws is 16B-aligned
    for (int k = lane; k < n4; k += 32) {
        float4 q = bs4[k];
        acc += (q.x + q.y) + (q.z + q.w);
    }
    for (int k = (n4 << 2) + lane; k < nblocks; k += 32) acc += blocksum[k];

    v2f a; a[0] = acc;  a[1] = 0.0f;   // A(16x4): lane m<16 -> A[m][0]; lane m>=16 -> A[m-16][2]
    v2f b; b[0] = 1.0f; b[1] = 1.0f;   // B(4x16) = ones
    v8f c = {0.0f, 0.0f, 0.0f, 0.0f, 0.0f, 0.0f, 0.0f, 0.0f};
    c = __builtin_amdgcn_wmma_f32_16x16x4_f32(
            /*neg_a=*/false, a, /*neg_b=*/false, b,
            /*c_mod=*/(short)0, c, /*reuse_a=*/false, /*reuse_b=*/false);

    // lanes 0..15 hold D[0..7][lane] in c[0..7]; lanes 16..31 hold D[8..15][lane-16]
    float s = c[0] + c[1] + c[2] + c[3] + c[4] + c[5] + c[6] + c[7];
    s += __shfl_xor(s, 16, 32);        // add upper-half rows -> full total in every lane
    if (lane == 0) out[0] = s * inv_batch;
}

extern "C" void kernel_launch(void* const* d_in, const int* in_sizes, int n_in,
                              void* d_out, int out_size, void* d_ws, size_t ws_size,
                              hipStream_t stream) {
    const float* pred = (const float*)d_in[0];   // [B,7,7,30] f32
    const float* gt   = (const float*)d_in[1];   // [B,7,7,25] f32
    float* out        = (float*)d_out;           // [1 + B*7*7*2] f32
    float* blocksum   = (float*)d_ws;            // nblocks f32 scratch

    const int ncells  = in_sizes[0] / 30;               // B*7*7 = 802816
    const int nblocks = (ncells + CPB - 1) / CPB;       // 3136 (exact: ncells % 256 == 0)
    const int batch   = ncells / 49;                    // 16384
    const float inv_batch = 1.0f / (float)batch;        // exact (power of two)

    yolo_cells<<<nblocks, 256, 0, stream>>>(pred, gt, out, blocksum, ncells);
    yolo_reduce<<<1, 32, 0, stream>>>(blocksum, nblocks, out, inv_batch);
}